// ControllerNetwork_4252017623044
// MI455X (gfx1250) — compile-verified
//
#include <hip/hip_runtime.h>
#include <hip/hip_bf16.h>
#include <math.h>

#define HIDDEN 2048
#define KSPLIT 2               // K-dimension split across waves (per gate)

typedef float v2f __attribute__((ext_vector_type(2)));
typedef float v8f __attribute__((ext_vector_type(8)));

// ---------------------------------------------------------------------------
// threefry2x32 (standard 20-round, JAX-compatible)
// ---------------------------------------------------------------------------
__device__ __forceinline__ unsigned rotl32(unsigned x, int r) {
    return (x << r) | (x >> (32 - r));
}

__device__ void threefry2x32(unsigned k0, unsigned k1, unsigned x0, unsigned x1,
                             unsigned* o0, unsigned* o1) {
    const unsigned ks2 = 0x1BD11BDAu ^ k0 ^ k1;
    unsigned v0 = x0 + k0, v1 = x1 + k1;
    const int ra[4] = {13, 15, 26, 6};
    const int rb[4] = {17, 29, 16, 24};
#pragma unroll
    for (int i = 0; i < 4; i++) { v0 += v1; v1 = rotl32(v1, ra[i]); v1 ^= v0; }
    v0 += k1; v1 += ks2 + 1u;
#pragma unroll
    for (int i = 0; i < 4; i++) { v0 += v1; v1 = rotl32(v1, rb[i]); v1 ^= v0; }
    v0 += ks2; v1 += k0 + 2u;
#pragma unroll
    for (int i = 0; i < 4; i++) { v0 += v1; v1 = rotl32(v1, ra[i]); v1 ^= v0; }
    v0 += k0; v1 += k1 + 3u;
#pragma unroll
    for (int i = 0; i < 4; i++) { v0 += v1; v1 = rotl32(v1, rb[i]); v1 ^= v0; }
    v0 += k1; v1 += ks2 + 4u;
#pragma unroll
    for (int i = 0; i < 4; i++) { v0 += v1; v1 = rotl32(v1, ra[i]); v1 ^= v0; }
    v0 += ks2; v1 += k0 + 5u;
    *o0 = v0; *o1 = v1;
}

__device__ __forceinline__ float gumbel_from_bits(unsigned bits) {
    float u = __uint_as_float((bits >> 9) | 0x3f800000u) - 1.0f;
    u = fmaxf(u, 1.1754944e-38f);
    return -logf(-logf(u));
}

__device__ __forceinline__ float sigmoidf_(float x) {
    return 1.0f / (1.0f + expf(-x));
}

// ---------------------------------------------------------------------------
// zero the h/c/x state in the workspace
// ---------------------------------------------------------------------------
__global__ void init_ws_kernel(float* ws, int n) {
    int i = blockIdx.x * blockDim.x + threadIdx.x;
    if (i < n) ws[i] = 0.0f;
}

// ---------------------------------------------------------------------------
// One LSTM layer. Block = 16 hidden units, 8 waves: wave w handles gate
// (w&3) and K-range half (w>>2). GEMV via V_WMMA_F32_16X16X4_F32 with the
// h-vector in column 0 of B. B fetch is an unconditional LDS broadcast load
// per half, zeroed for lanes m!=0 via cndmask (no exec masking).
// ---------------------------------------------------------------------------
__global__ __launch_bounds__(256) void lstm_layer_kernel(
    const float* __restrict__ Whh,   // (8192, 2048)
    const float* __restrict__ Wih,   // (8192, in_dim): in_dim=1 (layer0) or 2048
    const float* __restrict__ bih,   // (8192,)
    const float* __restrict__ bhh,   // (8192,)
    const float* __restrict__ hprev, // (2048,)  read (old parity)
    const float* __restrict__ cprev, // (2048,)  read (old parity)
    float* __restrict__ hout,        // (2048,)  write (new parity)
    float* __restrict__ cout,        // (2048,)  write (new parity)
    const float* __restrict__ vin,   // (2048,) layer-1 input vector, or null
    const float* __restrict__ xptr)  // scalar layer-0 input, or null
{
    __shared__ float sh_h[HIDDEN];
    __shared__ float sh_v[HIDDEN];
    __shared__ float sgate[4 * KSPLIT][16];

    const int t = threadIdx.x;
    for (int i = t; i < HIDDEN; i += 256) sh_h[i] = hprev[i];
    if (vin) {
        for (int i = t; i < HIDDEN; i += 256) sh_v[i] = vin[i];
    }
    __syncthreads();

    const int w      = t >> 5;           // wave id 0..7
    const int g      = w & 3;            // gate index (i,f,g,o)
    const int ksplit = w >> 2;           // which K half
    const int lane   = t & 31;
    const int m      = lane & 15;        // row within 16-row tile
    const int half   = lane >> 4;        // A/B: K = 2*half, 2*half+1
    const int j0     = blockIdx.x * 16;
    const bool bsel  = (m == 0);         // lanes 0/16 carry B column 0

    const int  kbeg    = ksplit * (HIDDEN / KSPLIT);
    const int  kend    = kbeg + (HIDDEN / KSPLIT);
    const long rowbase = (long)(g * HIDDEN + j0 + m);
    const v2f  zero2   = {0.f, 0.f};

    v8f acc0 = {0.f, 0.f, 0.f, 0.f, 0.f, 0.f, 0.f, 0.f};
    v8f acc1 = {0.f, 0.f, 0.f, 0.f, 0.f, 0.f, 0.f, 0.f};

    // ---- Whh @ hprev (this wave's K half) ----
    {
        const float* Arow = Whh + rowbase * HIDDEN + 2 * half;
        for (int kb = kbeg; kb < kend; kb += 8) {
            v2f a0  = *(const v2f*)(Arow + kb);
            v2f a1  = *(const v2f*)(Arow + kb + 4);
            v2f hh0 = *(const v2f*)(sh_h + kb + 2 * half);      // broadcast load
            v2f hh1 = *(const v2f*)(sh_h + kb + 4 + 2 * half);
            v2f b0  = bsel ? hh0 : zero2;                       // v_cndmask
            v2f b1  = bsel ? hh1 : zero2;
            acc0 = __builtin_amdgcn_wmma_f32_16x16x4_f32(false, a0, false, b0,
                                                         (short)0, acc0, false, false);
            acc1 = __builtin_amdgcn_wmma_f32_16x16x4_f32(false, a1, false, b1,
                                                         (short)0, acc1, false, false);
        }
    }
    // ---- Wih @ vin (layer 1 only, this wave's K half) ----
    if (vin) {
        const float* Arow = Wih + rowbase * HIDDEN + 2 * half;
        for (int kb = kbeg; kb < kend; kb += 8) {
            v2f a0  = *(const v2f*)(Arow + kb);
            v2f a1  = *(const v2f*)(Arow + kb + 4);
            v2f hh0 = *(const v2f*)(sh_v + kb + 2 * half);
            v2f hh1 = *(const v2f*)(sh_v + kb + 4 + 2 * half);
            v2f b0  = bsel ? hh0 : zero2;
            v2f b1  = bsel ? hh1 : zero2;
            acc0 = __builtin_amdgcn_wmma_f32_16x16x4_f32(false, a0, false, b0,
                                                         (short)0, acc0, false, false);
            acc1 = __builtin_amdgcn_wmma_f32_16x16x4_f32(false, a1, false, b1,
                                                         (short)0, acc1, false, false);
        }
    }

    v8f acc = acc0 + acc1;

    // D column N=0: lane 0 holds M=0..7 in v[0..7], lane 16 holds M=8..15.
    if (lane == 0) {
#pragma unroll
        for (int r = 0; r < 8; r++) sgate[w][r] = acc[r];
    } else if (lane == 16) {
#pragma unroll
        for (int r = 0; r < 8; r++) sgate[w][8 + r] = acc[r];
    }
    __syncthreads();

    // ---- fused cell update (PyTorch gate order i, f, g, o) ----
    if (t < 16) {
        const int j = j0 + t;
        const float x = xptr ? *xptr : 0.0f;
        float pre[4];
#pragma unroll
        for (int gg = 0; gg < 4; gg++) {
            const int row = gg * HIDDEN + j;
            float p = sgate[gg][t] + sgate[gg + 4][t] + bih[row] + bhh[row];
            if (xptr) p += Wih[row] * x;  // W_ih is (8192,1) for layer 0
            pre[gg] = p;
        }
        const float ig = sigmoidf_(pre[0]);
        const float fg = sigmoidf_(pre[1]);
        const float gt = tanhf(pre[2]);
        const float og = sigmoidf_(pre[3]);
        const float cn = fg * cprev[j] + ig * gt;
        cout[j] = cn;
        hout[j] = og * tanhf(cn);
    }
}

// ---------------------------------------------------------------------------
// logits = Wout @ h + bout ; categorical sample (JAX threefry + Gumbel argmax);
// write decision + prob, feed sampled index back as next-step x.
// ---------------------------------------------------------------------------
__global__ __launch_bounds__(32) void out_sample_kernel(
    const float* __restrict__ Wout, const float* __restrict__ bout,
    int n, int step, const float* __restrict__ h,
    float* __restrict__ x_ws, float* __restrict__ dout)
{
    __shared__ float slog[8];
    const int lane = threadIdx.x;

    for (int r = 0; r < n; r++) {
        float p = 0.0f;
        for (int k = lane; k < HIDDEN; k += 32) p += Wout[r * HIDDEN + k] * h[k];
#pragma unroll
        for (int off = 16; off > 0; off >>= 1) p += __shfl_down(p, off, 32);
        if (lane == 0) slog[r] = p + bout[r];
    }
    __syncthreads();

    if (lane == 0) {
        // folded key = threefry(key=(0,42), data=(0,step))
        unsigned f0, f1;
        threefry2x32(0u, 42u, 0u, (unsigned)step, &f0, &f1);

        // JAX counter-split random bits for n draws
        float gum[8];
        const int halfc = (n + 1) >> 1;
        for (int i = 0; i < halfc; i++) {
            const unsigned c0 = (unsigned)i;
            const unsigned c1 = (halfc + i < n) ? (unsigned)(halfc + i) : 0u;
            unsigned o0, o1;
            threefry2x32(f0, f1, c0, c1, &o0, &o1);
            gum[i] = gumbel_from_bits(o0);
            if (halfc + i < n) gum[halfc + i] = gumbel_from_bits(o1);
        }

        int   idx  = 0;
        float best = slog[0] + gum[0];
        for (int i = 1; i < n; i++) {
            const float z = slog[i] + gum[i];
            if (z > best) { best = z; idx = i; }
        }

        float mx = slog[0];
        for (int i = 1; i < n; i++) mx = fmaxf(mx, slog[i]);
        float s = 0.0f;
        for (int i = 0; i < n; i++) s += expf(slog[i] - mx);
        const float prob = expf(slog[idx] - mx) / s;

        dout[step]     = (float)idx;   // decisions
        dout[3 + step] = prob;         // probs
        *x_ws          = (float)idx;   // feedback token
    }
}

// ---------------------------------------------------------------------------
// kernel_launch
// ---------------------------------------------------------------------------
extern "C" void kernel_launch(void* const* d_in, const int* in_sizes, int n_in,
                              void* d_out, int out_size, void* d_ws, size_t ws_size,
                              hipStream_t stream) {
    // JAX pytree flatten order (sorted dict keys):
    // 0: inutile
    // 1: W_hh[0]  2: W_hh[1]  3: W_ih[0]  4: W_ih[1]
    // 5: W_out[0] 6: W_out[1] 7: W_out[2]
    // 8: b_hh[0]  9: b_hh[1] 10: b_ih[0] 11: b_ih[1]
    // 12: b_out[0] 13: b_out[1] 14: b_out[2]
    const float* Whh[2]  = {(const float*)d_in[1],  (const float*)d_in[2]};
    const float* Wih[2]  = {(const float*)d_in[3],  (const float*)d_in[4]};
    const float* Wout[3] = {(const float*)d_in[5],  (const float*)d_in[6],  (const float*)d_in[7]};
    const float* bhh[2]  = {(const float*)d_in[8],  (const float*)d_in[9]};
    const float* bih[2]  = {(const float*)d_in[10], (const float*)d_in[11]};
    const float* bout[3] = {(const float*)d_in[12], (const float*)d_in[13], (const float*)d_in[14]};

    float* ws = (float*)d_ws;
    // ws layout (floats): H[layer][parity][2048] | C[layer][parity][2048] | x
    float* H = ws;                 // 4 * 2048
    float* C = ws + 4 * HIDDEN;    // 4 * 2048
    float* X = ws + 8 * HIDDEN;    // 1
    const int ws_used = 8 * HIDDEN + 1;

    init_ws_kernel<<<(ws_used + 255) / 256, 256, 0, stream>>>(ws, ws_used);

    const int nch[3] = {6, 4, 3};
    for (int s = 0; s < 3; s++) {
        const int rp = s & 1;       // read parity
        const int wp = rp ^ 1;      // write parity
        // layer 0: scalar input x
        lstm_layer_kernel<<<HIDDEN / 16, 256, 0, stream>>>(
            Whh[0], Wih[0], bih[0], bhh[0],
            H + (0 * 2 + rp) * HIDDEN, C + (0 * 2 + rp) * HIDDEN,
            H + (0 * 2 + wp) * HIDDEN, C + (0 * 2 + wp) * HIDDEN,
            nullptr, X);
        // layer 1: input = new h of layer 0
        lstm_layer_kernel<<<HIDDEN / 16, 256, 0, stream>>>(
            Whh[1], Wih[1], bih[1], bhh[1],
            H + (1 * 2 + rp) * HIDDEN, C + (1 * 2 + rp) * HIDDEN,
            H + (1 * 2 + wp) * HIDDEN, C + (1 * 2 + wp) * HIDDEN,
            H + (0 * 2 + wp) * HIDDEN, nullptr);
        // output head + sample
        out_sample_kernel<<<1, 32, 0, stream>>>(
            Wout[s], bout[s], nch[s], s,
            H + (1 * 2 + wp) * HIDDEN, X, (float*)d_out);
    }
}